// PinSAGEHetero_2095944040611
// MI455X (gfx1250) — compile-verified
//
#include <hip/hip_runtime.h>
#include <hip/hip_bf16.h>

// ---------------------------------------------------------------------------
// PinSAGE-hetero forward on gfx1250.
// GEMMs via V_WMMA_F32_16X16X4_F32 (exact fp32 path, 16x16 tile per wave32).
// Segment-mean via global_atomic_add_f32 scatter (L2-resident destinations).
// ---------------------------------------------------------------------------

typedef float v2f __attribute__((ext_vector_type(2)));
typedef float v8f __attribute__((ext_vector_type(8)));

#define N_USER 100000
#define N_PROB 20000
#define N_EDGE 600000
#define HDIM   128
#define ODIM   64
#define U_IN   256
#define P_IN   128

// ---------------------------------------------------------------------------
// Fused GEMM:  Y[M,N] = maybe_relu( X1[M,K1] @ W1[N,K1]^T + bias[N]
//                                 (+ X2[M,K2] @ W2[N,K2]^T  if X2 != null) )
// One wave computes a 16-row x N strip. NT = N/16 accumulator tiles in VGPRs.
// Requires M % 16 == 0 (true for all call sites) and K % 4 == 0.
// ---------------------------------------------------------------------------
template <int NT>
__global__ __launch_bounds__(128) void gemm_dual_wmma(
    const float* __restrict__ X1, const float* __restrict__ W1,
    const float* __restrict__ bias,
    const float* __restrict__ X2, const float* __restrict__ W2,
    float* __restrict__ Y, int M, int K1, int K2, int do_relu) {
  constexpr int N = NT * 16;
  const int lane  = threadIdx.x & 31;
  const int wid   = threadIdx.x >> 5;
  const int mTile = blockIdx.x * (blockDim.x >> 5) + wid;  // wave-uniform
  if (mTile * 16 >= M) return;                             // whole wave exits
  const int m0  = mTile * 16;
  const int row = lane & 15;   // A: M row / B: N col / C: N col
  const int kh  = lane >> 4;   // selects K pair (A/B) and M half (C/D)

  v8f acc[NT];
#pragma unroll
  for (int nt = 0; nt < NT; ++nt) {
    const float bv = bias[nt * 16 + row];  // depends on N column only
#pragma unroll
    for (int r = 0; r < 8; ++r) acc[nt][r] = bv;
  }

  // --- pass 1: X1 @ W1^T ---
  {
    const float* xr = X1 + (size_t)(m0 + row) * K1;
    for (int k0 = 0; k0 < K1; k0 += 4) {
      const int ka = k0 + kh * 2;
      v2f a;
      a[0] = xr[ka];
      a[1] = xr[ka + 1];
#pragma unroll
      for (int nt = 0; nt < NT; ++nt) {
        const float* wr = W1 + (size_t)(nt * 16 + row) * K1;
        v2f b;
        b[0] = wr[ka];
        b[1] = wr[ka + 1];
        acc[nt] = __builtin_amdgcn_wmma_f32_16x16x4_f32(
            false, a, false, b, (short)0, acc[nt], false, false);
      }
    }
  }
  // --- pass 2 (optional): X2 @ W2^T ---
  if (X2 != nullptr) {
    const float* xr = X2 + (size_t)(m0 + row) * K2;
    for (int k0 = 0; k0 < K2; k0 += 4) {
      const int ka = k0 + kh * 2;
      v2f a;
      a[0] = xr[ka];
      a[1] = xr[ka + 1];
#pragma unroll
      for (int nt = 0; nt < NT; ++nt) {
        const float* wr = W2 + (size_t)(nt * 16 + row) * K2;
        v2f b;
        b[0] = wr[ka];
        b[1] = wr[ka + 1];
        acc[nt] = __builtin_amdgcn_wmma_f32_16x16x4_f32(
            false, a, false, b, (short)0, acc[nt], false, false);
      }
    }
  }

  // --- store: C/D layout -> Y[m0 + r + kh*8, nt*16 + row] ---
#pragma unroll
  for (int nt = 0; nt < NT; ++nt) {
#pragma unroll
    for (int r = 0; r < 8; ++r) {
      float v = acc[nt][r];
      if (do_relu) v = fmaxf(v, 0.0f);
      Y[(size_t)(m0 + r + kh * 8) * N + nt * 16 + row] = v;
    }
  }
}

// ---------------------------------------------------------------------------
// Edge scatter-add: 128 threads per edge (one per feature element).
// acc[dst[e]][f] += feat[src[e]][f]  via fp32 global atomic add.
// ---------------------------------------------------------------------------
__global__ __launch_bounds__(256) void scatter_add_k(
    const float* __restrict__ feat, const int* __restrict__ src,
    const int* __restrict__ dst, int nE, float* __restrict__ acc) {
  const int f = threadIdx.x & 127;
  const int e = blockIdx.x * (blockDim.x >> 7) + (threadIdx.x >> 7);
  if (e >= nE) return;
  const int s = src[e];
  const int d = dst[e];
  const float v = feat[(size_t)s * HDIM + f];
  __hip_atomic_fetch_add(&acc[(size_t)d * HDIM + f], v, __ATOMIC_RELAXED,
                         __HIP_MEMORY_SCOPE_AGENT);
}

__global__ __launch_bounds__(256) void count_edges_k(
    const int* __restrict__ dst, int nE, float* __restrict__ cnt) {
  const int e = blockIdx.x * blockDim.x + threadIdx.x;
  if (e >= nE) return;
  __hip_atomic_fetch_add(&cnt[dst[e]], 1.0f, __ATOMIC_RELAXED,
                         __HIP_MEMORY_SCOPE_AGENT);
}

__global__ __launch_bounds__(256) void recip_k(float* __restrict__ cnt, int n) {
  const int i = blockIdx.x * blockDim.x + threadIdx.x;
  if (i < n) cnt[i] = 1.0f / fmaxf(cnt[i], 1.0f);
}

// acc[i] *= rcnt[i / 128]   (i over rows*128 elements)
__global__ __launch_bounds__(256) void scale_rows_k(
    float* __restrict__ acc, const float* __restrict__ rcnt, int nElem) {
  const int i = blockIdx.x * blockDim.x + threadIdx.x;
  if (i < nElem) acc[i] *= rcnt[i >> 7];
}

// ---------------------------------------------------------------------------
extern "C" void kernel_launch(void* const* d_in, const int* in_sizes, int n_in,
                              void* d_out, int out_size, void* d_ws,
                              size_t ws_size, hipStream_t stream) {
  (void)in_sizes; (void)n_in; (void)out_size; (void)ws_size;

  const float* x_user = (const float*)d_in[0];
  const float* x_prob = (const float*)d_in[1];
  const int*   e_up_s = (const int*)d_in[2];
  const int*   e_up_d = (const int*)d_in[3];
  const int*   e_pu_s = (const int*)d_in[4];
  const int*   e_pu_d = (const int*)d_in[5];
  const float* Wu     = (const float*)d_in[6];
  const float* bu     = (const float*)d_in[7];
  const float* Wp     = (const float*)d_in[8];
  const float* bp     = (const float*)d_in[9];
  const float* Wl_up  = (const float*)d_in[10];
  const float* bl_up  = (const float*)d_in[11];
  const float* Wr_up  = (const float*)d_in[12];
  const float* Wl_pu  = (const float*)d_in[13];
  const float* bl_pu  = (const float*)d_in[14];
  const float* Wr_pu  = (const float*)d_in[15];
  const float* Wou    = (const float*)d_in[16];
  const float* bou    = (const float*)d_in[17];
  const float* Wop    = (const float*)d_in[18];
  const float* bop    = (const float*)d_in[19];

  // ---- workspace carve-out ----
  char*  ws  = (char*)d_ws;
  size_t off = 0;
  auto carve = [&](size_t bytes) -> float* {
    float* p = (float*)(ws + off);
    off = (off + bytes + 255) & ~(size_t)255;
    return p;
  };
  float* u0    = carve((size_t)N_USER * HDIM * 4);
  float* u1    = carve((size_t)N_USER * HDIM * 4);
  float* p0    = carve((size_t)N_PROB * HDIM * 4);
  float* p1    = carve((size_t)N_PROB * HDIM * 4);
  float* agg_u = carve((size_t)N_USER * HDIM * 4);
  float* agg_p = carve((size_t)N_PROB * HDIM * 4);
  float* rc_u  = carve((size_t)N_USER * 4);
  float* rc_p  = carve((size_t)N_PROB * 4);

  const dim3 blk(128);
  auto mblocks = [](int M) { return ((M / 16) + 3) / 4; };  // 4 waves / block

  // ---- input projections ----
  gemm_dual_wmma<8><<<mblocks(N_USER), blk, 0, stream>>>(
      x_user, Wu, bu, nullptr, nullptr, u0, N_USER, U_IN, 0, 0);
  gemm_dual_wmma<8><<<mblocks(N_PROB), blk, 0, stream>>>(
      x_prob, Wp, bp, nullptr, nullptr, p0, N_PROB, P_IN, 0, 0);

  // ---- in-degree reciprocals (loop-invariant) ----
  hipMemsetAsync(rc_p, 0, (size_t)N_PROB * 4, stream);
  hipMemsetAsync(rc_u, 0, (size_t)N_USER * 4, stream);
  count_edges_k<<<(N_EDGE + 255) / 256, 256, 0, stream>>>(e_up_d, N_EDGE, rc_p);
  count_edges_k<<<(N_EDGE + 255) / 256, 256, 0, stream>>>(e_pu_d, N_EDGE, rc_u);
  recip_k<<<(N_PROB + 255) / 256, 256, 0, stream>>>(rc_p, N_PROB);
  recip_k<<<(N_USER + 255) / 256, 256, 0, stream>>>(rc_u, N_USER);

  // ---- 2 SAGE layers ----
  float* ucur = u0; float* unxt = u1;
  float* pcur = p0; float* pnxt = p1;
  const int edgeBlocks = (N_EDGE + 1) / 2;  // 2 edges per 256-thread block
  for (int i = 0; i < 2; ++i) {
    // user -> problem mean aggregation
    hipMemsetAsync(agg_p, 0, (size_t)N_PROB * HDIM * 4, stream);
    scatter_add_k<<<edgeBlocks, 256, 0, stream>>>(ucur, e_up_s, e_up_d, N_EDGE, agg_p);
    scale_rows_k<<<(N_PROB * HDIM + 255) / 256, 256, 0, stream>>>(
        agg_p, rc_p, N_PROB * HDIM);
    // problem -> user mean aggregation
    hipMemsetAsync(agg_u, 0, (size_t)N_USER * HDIM * 4, stream);
    scatter_add_k<<<edgeBlocks, 256, 0, stream>>>(pcur, e_pu_s, e_pu_d, N_EDGE, agg_u);
    scale_rows_k<<<(N_USER * HDIM + 255) / 256, 256, 0, stream>>>(
        agg_u, rc_u, N_USER * HDIM);

    const float* Wl1 = Wl_up + (size_t)i * HDIM * HDIM;
    const float* bl1 = bl_up + (size_t)i * HDIM;
    const float* Wr1 = Wr_up + (size_t)i * HDIM * HDIM;
    const float* Wl2 = Wl_pu + (size_t)i * HDIM * HDIM;
    const float* bl2 = bl_pu + (size_t)i * HDIM;
    const float* Wr2 = Wr_pu + (size_t)i * HDIM * HDIM;

    // p_new = relu(agg_p @ Wl_up^T + bl_up + p @ Wr_up^T)
    gemm_dual_wmma<8><<<mblocks(N_PROB), blk, 0, stream>>>(
        agg_p, Wl1, bl1, pcur, Wr1, pnxt, N_PROB, HDIM, HDIM, 1);
    // u_new = relu(agg_u @ Wl_pu^T + bl_pu + u @ Wr_pu^T)
    gemm_dual_wmma<8><<<mblocks(N_USER), blk, 0, stream>>>(
        agg_u, Wl2, bl2, ucur, Wr2, unxt, N_USER, HDIM, HDIM, 1);

    float* t;
    t = ucur; ucur = unxt; unxt = t;
    t = pcur; pcur = pnxt; pnxt = t;
  }

  // ---- output heads ----
  float* out_u = (float*)d_out;
  float* out_p = out_u + (size_t)N_USER * ODIM;
  gemm_dual_wmma<4><<<mblocks(N_USER), blk, 0, stream>>>(
      ucur, Wou, bou, nullptr, nullptr, out_u, N_USER, HDIM, 0, 0);
  gemm_dual_wmma<4><<<mblocks(N_PROB), blk, 0, stream>>>(
      pcur, Wop, bop, nullptr, nullptr, out_p, N_PROB, HDIM, 0, 0);
}